// PointSelector_51470888075602
// MI455X (gfx1250) — compile-verified
//
#include <hip/hip_runtime.h>
#include <hip/hip_bf16.h>
#include <math.h>

// Problem constants (from reference)
static constexpr int Bb = 4;
static constexpr int Nn = 16384;
static constexpr int Cc = 512;
static constexpr int Mm = 512;
static constexpr int Kk = 16;
static constexpr int WL = 8;
static constexpr float TEMP = 0.1f;

typedef __attribute__((ext_vector_type(2))) float v2f;
typedef __attribute__((ext_vector_type(8))) float v8f;

#define ROWS (Bb * Mm) // 2048 flattened walkers

// ---------------------------------------------------------------------------
// Init: cur = start_indices; out[:, :, 0, :] = feats[start]
// ---------------------------------------------------------------------------
__global__ void ps_init_kernel(const float* __restrict__ feats,
                               const int* __restrict__ start,
                               int* __restrict__ cur,
                               float* __restrict__ out) {
    int bid = blockIdx.x;            // walker id in [0, ROWS)
    int tid = threadIdx.x;           // 256 threads
    int b = bid / Mm;
    int si = start[bid];
    if (tid == 0) cur[bid] = si;
    const float* fr = feats + ((size_t)b * Nn + si) * Cc;
    float* orow = out + ((size_t)bid * WL + 0) * Cc;
    for (int c = tid; c < Cc; c += 256) orow[c] = fr[c];
}

// ---------------------------------------------------------------------------
// KNN: for each walker, 17 nearest points (incl. self); write ranks 1..16.
// Coords are streamed through LDS in 24KB tiles using CDNA5 async
// global->LDS copies (ASYNCcnt), double-buffered against the distance loop.
// Per-thread sorted top-17, then 17 rounds of LDS argmin merge.
// Tie-break: smaller point index wins (matches jax.lax.top_k stability).
// ---------------------------------------------------------------------------
static constexpr int TPTS = 2048;            // points per LDS tile
static constexpr int NTILES = Nn / TPTS;     // 8
static constexpr int TBYTES = TPTS * 3 * 4;  // 24576 bytes per tile

__global__ __launch_bounds__(256) void ps_knn_kernel(
    const float* __restrict__ coords, const int* __restrict__ cur,
    int* __restrict__ knn) {
    const int bid = blockIdx.x;      // walker
    const int tid = threadIdx.x;
    const int b = bid / Mm;
    const int ci = cur[bid];
    const float* cb = coords + (size_t)b * Nn * 3;
    const float ccx = cb[(size_t)ci * 3 + 0];
    const float ccy = cb[(size_t)ci * 3 + 1];
    const float ccz = cb[(size_t)ci * 3 + 2];

    __shared__ float sc[2][TPTS * 3];    // 2 x 24KB coord tiles
    __shared__ float sd[256];
    __shared__ int sp[256];

    const unsigned ldsb[2] = {(unsigned)(uintptr_t)(&sc[0][0]),
                              (unsigned)(uintptr_t)(&sc[1][0])};

    // Issue one tile's async copies: 256 threads x 6 x b128 = 24KB.
    auto issue_tile = [&](int tile, unsigned lbase) {
        const char* g = (const char*)(cb + (size_t)tile * TPTS * 3) + tid * 16;
        unsigned l = lbase + tid * 16;
#pragma unroll
        for (int i = 0; i < 6; ++i) {
            asm volatile("global_load_async_to_lds_b128 %0, %1, off"
                         :: "v"(l + i * 4096), "v"(g + (size_t)i * 4096)
                         : "memory");
        }
    };

    const float INF = __builtin_inff();
    float dl[17];
    int il[17];
#pragma unroll
    for (int j = 0; j < 17; ++j) { dl[j] = INF; il[j] = 0x3FFF; }

    issue_tile(0, ldsb[0]);
    asm volatile("s_wait_asynccnt 0x0" ::: "memory");
    __syncthreads();

    for (int t = 0; t < NTILES; ++t) {
        if (t + 1 < NTILES) issue_tile(t + 1, ldsb[(t + 1) & 1]);
        const float* scb = sc[t & 1];
        for (int pi = tid; pi < TPTS; pi += 256) {
            float dx = scb[pi * 3 + 0] - ccx;
            float dy = scb[pi * 3 + 1] - ccy;
            float dz = scb[pi * 3 + 2] - ccz;
            float d2 = dx * dx + dy * dy + dz * dz;
            int idx = t * TPTS + pi;
            // fast reject vs current worst, then bubble-insert (ascending)
            if (d2 < dl[16] || (d2 == dl[16] && idx < il[16])) {
                float cd = d2;
                int cidx = idx;
#pragma unroll
                for (int j = 0; j < 17; ++j) {
                    bool sw = (cd < dl[j]) || (cd == dl[j] && cidx < il[j]);
                    float td = sw ? dl[j] : cd;
                    int ti = sw ? il[j] : cidx;
                    dl[j] = sw ? cd : dl[j];
                    il[j] = sw ? cidx : il[j];
                    cd = td;
                    cidx = ti;
                }
            }
        }
        asm volatile("s_wait_asynccnt 0x0" ::: "memory");
        __syncthreads();
    }

    // Block-wide 17-round argmin merge.
    for (int r = 0; r < 17; ++r) {
        sd[tid] = dl[0];
        sp[tid] = (il[0] & 0x3FFF) | (tid << 14); // idx low 14 bits, tid above
        __syncthreads();
        for (int off = 128; off > 0; off >>= 1) {
            if (tid < off) {
                float dv = sd[tid + off];
                int pv = sp[tid + off];
                bool take = (dv < sd[tid]) ||
                            (dv == sd[tid] && (pv & 0x3FFF) < (sp[tid] & 0x3FFF));
                if (take) { sd[tid] = dv; sp[tid] = pv; }
            }
            __syncthreads();
        }
        int wp = sp[0];
        int widx = wp & 0x3FFF;
        int wtid = wp >> 14;
        if (r >= 1 && tid == 0) knn[(size_t)bid * Kk + (r - 1)] = widx;
        if (tid == wtid) {
            // pop head of local sorted list (static-index shift)
#pragma unroll
            for (int j = 0; j < 16; ++j) { dl[j] = dl[j + 1]; il[j] = il[j + 1]; }
            dl[16] = INF;
            il[16] = 0x3FFF;
        }
        __syncthreads();
    }
}

// ---------------------------------------------------------------------------
// WMMA f32 GEMM: out[m][n] = sum_k A[m][k] * B[k][n] (+ bias[n]).
// One wave computes a 16x64 tile: 4 independent accumulator chains sharing
// each A fragment (V_WMMA_F32_16X16X4_F32, K-chunks of 4).
// GATHER: A rows are feats[cur[m]]. TRANSB: B[k][n] = W[n*C + k] (contig b64).
// ---------------------------------------------------------------------------
template <bool GATHER, bool TRANSB>
__global__ __launch_bounds__(32) void ps_gemm16_kernel(
    const float* __restrict__ src,   // GATHER ? feats : A (ROWS x C)
    const int* __restrict__ gidx,    // GATHER ? cur : unused
    const float* __restrict__ W,     // C x C
    const float* __restrict__ bias,  // C or nullptr
    float* __restrict__ outp) {      // ROWS x C
    const int lane = threadIdx.x;
    const int m0 = blockIdx.y * 16;
    const int n0 = blockIdx.x * 64;
    const int lrow = lane & 15;
    const int hi = lane >> 4;
    const int koff = hi * 2;

    const float* arow;
    if (GATHER) {
        int r = m0 + lrow;
        int b = r / Mm;
        int idx = gidx[r];
        arow = src + ((size_t)b * Nn + idx) * Cc;
    } else {
        arow = src + (size_t)(m0 + lrow) * Cc;
    }
    // Prefetch the (gathered) 2KB A row — dominant HBM traffic is this gather.
#pragma unroll
    for (int pf = 0; pf < Cc; pf += 64) __builtin_prefetch(arow + pf, 0, 1);

    v8f acc[4] = {{}, {}, {}, {}};
#if defined(__has_builtin) && __has_builtin(__builtin_amdgcn_wmma_f32_16x16x4_f32)
    const float* brow[4];
    if (TRANSB) {
#pragma unroll
        for (int t = 0; t < 4; ++t)
            brow[t] = W + (size_t)(n0 + t * 16 + lrow) * Cc;
    }
    for (int k = 0; k < Cc; k += 4) {
        v2f a;
        a.x = arow[k + koff];
        a.y = arow[k + koff + 1];
#pragma unroll
        for (int t = 0; t < 4; ++t) {
            v2f bv;
            if (TRANSB) {
                bv.x = brow[t][k + koff];
                bv.y = brow[t][k + koff + 1];
            } else {
                bv.x = W[(size_t)(k + koff) * Cc + n0 + t * 16 + lrow];
                bv.y = W[(size_t)(k + koff + 1) * Cc + n0 + t * 16 + lrow];
            }
            acc[t] = __builtin_amdgcn_wmma_f32_16x16x4_f32(
                false, a, false, bv, (short)0, acc[t], false, false);
        }
    }
#else
    // Scalar fallback (correct, slow): each lane owns its 8 rows x 4 cols.
    const float* rp[8];
#pragma unroll
    for (int r = 0; r < 8; ++r) {
        int row = m0 + hi * 8 + r;
        if (GATHER) {
            int b = row / Mm;
            rp[r] = src + ((size_t)b * Nn + gidx[row]) * Cc;
        } else {
            rp[r] = src + (size_t)row * Cc;
        }
    }
    for (int k = 0; k < Cc; ++k) {
#pragma unroll
        for (int t = 0; t < 4; ++t) {
            int col = n0 + t * 16 + lrow;
            float bk = TRANSB ? W[(size_t)col * Cc + k] : W[(size_t)k * Cc + col];
#pragma unroll
            for (int r = 0; r < 8; ++r) acc[t][r] += rp[r][k] * bk;
        }
    }
#endif

#pragma unroll
    for (int t = 0; t < 4; ++t) {
        int col = n0 + t * 16 + lrow;
        float bcol = bias ? bias[col] : 0.0f;
#pragma unroll
        for (int r = 0; r < 8; ++r) {
            outp[(size_t)(m0 + hi * 8 + r) * Cc + col] = acc[t][r] + bcol;
        }
    }
}

// ---------------------------------------------------------------------------
// Select: logits_k = (f_nb_k . p + q . bg) / sqrt(C); Gumbel argmax; advance
// cur; write feats[next] into out[:, :, s+1, :].
// ---------------------------------------------------------------------------
__global__ __launch_bounds__(256) void ps_select_kernel(
    const float* __restrict__ feats, const float* __restrict__ q,
    const float* __restrict__ p, const int* __restrict__ knn,
    const float* __restrict__ bg, const float* __restrict__ gu, // step slice
    int* __restrict__ cur, float* __restrict__ out, int step) {
    const int bid = blockIdx.x;
    const int tid = threadIdx.x;
    const int b = bid / Mm;
    const float inv_sqrt_c = 0.044194173824159216f; // 1/sqrt(512)

    const float* qr = q + (size_t)bid * Cc;
    const float* pr = p + (size_t)bid * Cc;

    __shared__ float sred[256];
    __shared__ float slog[Kk];
    __shared__ int snxt;

    // Prefetch gathered neighbor row early (each 16-thread group owns one).
    const int k = tid >> 4;
    const int lg = tid & 15;
    const int nb = knn[(size_t)bid * Kk + k];
    const float* fr = feats + ((size_t)b * Nn + nb) * Cc;
    if (lg < 8) __builtin_prefetch(fr + lg * 64, 0, 1);

    // q . bg (block reduction)
    float part = 0.0f;
    for (int c = tid; c < Cc; c += 256) part += qr[c] * bg[c];
    sred[tid] = part;
    __syncthreads();
    for (int off = 128; off > 0; off >>= 1) {
        if (tid < off) sred[tid] += sred[tid + off];
        __syncthreads();
    }

    // neighbor dots: 16 groups of 16 threads
    float dp = 0.0f;
    for (int c = lg; c < Cc; c += 16) dp += fr[c] * pr[c];
#pragma unroll
    for (int off = 8; off > 0; off >>= 1) dp += __shfl_down(dp, off, 16);
    if (lg == 0) slog[k] = dp;
    __syncthreads();

    if (tid == 0) {
        const float qbg = sred[0];
        float best = -__builtin_inff();
        int bestk = 0;
        for (int kk2 = 0; kk2 < Kk; ++kk2) {
            float logit = (slog[kk2] + qbg) * inv_sqrt_c;
            float u = gu[(size_t)bid * Kk + kk2];
            float g = -logf(-logf(u));
            float sc = (logit + g) / TEMP;
            if (sc > best) { best = sc; bestk = kk2; }
        }
        int nx = knn[(size_t)bid * Kk + bestk];
        cur[bid] = nx;
        snxt = nx;
    }
    __syncthreads();

    const int nx = snxt;
    const float* fo = feats + ((size_t)b * Nn + nx) * Cc;
    float* orow = out + ((size_t)bid * WL + (step + 1)) * Cc;
    for (int c = tid; c < Cc; c += 256) orow[c] = fo[c];
}

// ---------------------------------------------------------------------------
extern "C" void kernel_launch(void* const* d_in, const int* in_sizes, int n_in,
                              void* d_out, int out_size, void* d_ws,
                              size_t ws_size, hipStream_t stream) {
    const float* feats = (const float*)d_in[0];
    const float* coords = (const float*)d_in[1];
    const int* start = (const int*)d_in[2];
    const float* Wb = (const float*)d_in[3];
    const float* bb = (const float*)d_in[4];
    const float* Wg = (const float*)d_in[5];
    const float* bg = (const float*)d_in[6];
    const float* gu = (const float*)d_in[7]; // (7, B, M, K)
    float* out = (float*)d_out;

    char* ws = (char*)d_ws;
    int* cur = (int*)ws;                                  //   8 KB
    int* knn = (int*)(ws + (16 << 10));                   // 128 KB
    float* q = (float*)(ws + (16 << 10) + (128 << 10));   //   4 MB
    float* p = (float*)(ws + (16 << 10) + (128 << 10) + ((size_t)ROWS * Cc * 4));

    ps_init_kernel<<<ROWS, 256, 0, stream>>>(feats, start, cur, out);

    const dim3 ggrid(Cc / 64, ROWS / 16); // (8, 128) 16x64 tiles
    for (int s = 0; s < WL - 1; ++s) {
        ps_knn_kernel<<<ROWS, 256, 0, stream>>>(coords, cur, knn);
        ps_gemm16_kernel<true, false><<<ggrid, 32, 0, stream>>>(
            feats, cur, Wb, bb, q);
        ps_gemm16_kernel<false, true><<<ggrid, 32, 0, stream>>>(
            q, nullptr, Wg, nullptr, p);
        ps_select_kernel<<<ROWS, 256, 0, stream>>>(
            feats, q, p, knn, bg, gu + (size_t)s * ROWS * Kk, cur, out, s);
    }
}